// EGCL_A2V_31928786878565
// MI455X (gfx1250) — compile-verified
//
#include <hip/hip_runtime.h>
#include <math.h>

// ---------------------------------------------------------------------------
// EGCL aggregate-to-virtual layer, fp32 WMMA (v_wmma_f32_16x16x4_f32) on
// gfx1250. N=200000 nodes, B=100 graphs, H=NF=128, C=3 virtual channels.
// Round 2: fragment-ordered packed weights (1 coalesced b64 load / wmma),
// A-fragment hoisting across the 8 N-tiles (8 live accumulators), and
// async global->LDS staging of the node_feat tile (ASYNCcnt path).
// ---------------------------------------------------------------------------

#define NNODES 200000
#define NGRAPH 100
#define HDIM   128
#define NCH    3
#define NFEAT  128
#define TILE   16
#define PITCH  132              // LDS row pitch in floats (conflict avoidance)
#define NTILES (NNODES / TILE)  // 12500 exactly
#define WAVES_PER_BLOCK 8
#define WAVE_LDS_FLOATS 6432    // 3*16*132 + 96

typedef float v2f __attribute__((ext_vector_type(2)));
typedef float v8f __attribute__((ext_vector_type(8)));

#define LDS_FENCE() asm volatile("s_wait_dscnt 0" ::: "memory")

__device__ __forceinline__ float silu_f(float x) {
    return x / (1.0f + __expf(-x));
}

// --------------------------- tiny helper kernels ---------------------------

__global__ void k_zero(float* p, int n) {
    int i = blockIdx.x * blockDim.x + threadIdx.x;
    if (i < n) p[i] = 0.0f;
}

// Pack the three 128x128 weight matrices into WMMA B-fragment order:
// Wp[((mat*8 + j)*32 + kc)*32 + lane] = { W[4kc+2h+0][16j+l16], W[4kc+2h+1][16j+l16] }
__global__ void k_pack(const float* __restrict__ We1,   // rows 0..127 used
                       const float* __restrict__ We2,
                       const float* __restrict__ Wc1,
                       float* __restrict__ Wp) {
    int t = blockIdx.x * blockDim.x + threadIdx.x;      // 0 .. 3*8*32*32-1
    if (t >= 3 * 8 * 32 * 32) return;
    int lane = t & 31;
    int kc   = (t >> 5) & 31;
    int j    = (t >> 10) & 7;
    int mat  = t >> 13;
    const float* W = (mat == 0) ? We1 : (mat == 1) ? We2 : Wc1;
    int half = lane >> 4, l16 = lane & 15;
    int k0  = kc * 4 + half * 2;
    int col = j * 16 + l16;
    float2 v = make_float2(W[k0 * HDIM + col], W[(k0 + 1) * HDIM + col]);
    *(float2*)(Wp + (size_t)t * 2) = v;
}

// vbias[b,c,n] = be1[n] + sum_k vnf[b,k,c] * We1[128+k, n]
__global__ void k_vbias(const float* __restrict__ vnf,
                        const float* __restrict__ We1,
                        const float* __restrict__ be1,
                        float* __restrict__ vbias) {
    int bc = blockIdx.x;            // 0..B*C-1
    int b = bc / NCH, c = bc % NCH;
    int n = threadIdx.x;            // 0..127
    float acc = be1[n];
    for (int k = 0; k < NFEAT; ++k)
        acc += vnf[(b * NFEAT + k) * NCH + c] * We1[(NFEAT + k) * HDIM + n];
    vbias[bc * HDIM + n] = acc;
}

__global__ void k_counts(const int* __restrict__ batch, float* __restrict__ counts) {
    int i = blockIdx.x * blockDim.x + threadIdx.x;
    if (i < NNODES) atomicAdd(&counts[batch[i]], 1.0f);
}

// out_coord = virtual_coord + transSum / max(count,1)
__global__ void k_final_coord(const float* __restrict__ vcoord,
                              const float* __restrict__ transSum,
                              const float* __restrict__ counts,
                              float* __restrict__ out) {
    int i = blockIdx.x * blockDim.x + threadIdx.x;
    if (i < NGRAPH * 3 * NCH) {
        int b = i / (3 * NCH);
        out[i] = vcoord[i] + transSum[i] / fmaxf(counts[b], 1.0f);
    }
}

// virtual node MLP: 300 rows, one block per (b,c)
__global__ void k_node_mlp(const float* __restrict__ vnf,
                           const float* __restrict__ aggSum,
                           const float* __restrict__ counts,
                           const float* __restrict__ Wn1,
                           const float* __restrict__ bn1,
                           const float* __restrict__ Wn2,
                           const float* __restrict__ bn2,
                           float* __restrict__ out) {
    __shared__ float hid[HDIM];
    int bc = blockIdx.x;
    int b = bc / NCH, c = bc % NCH;
    int n = threadIdx.x;
    float cnt = fmaxf(counts[b], 1.0f);
    float acc = bn1[n];
    for (int k = 0; k < NFEAT; ++k)
        acc += vnf[(b * NFEAT + k) * NCH + c] * Wn1[k * HDIM + n];
    for (int k = 0; k < HDIM; ++k)
        acc += (aggSum[bc * HDIM + k] / cnt) * Wn1[(NFEAT + k) * HDIM + n];
    hid[n] = silu_f(acc);
    __syncthreads();
    float o = bn2[n];
    for (int k = 0; k < HDIM; ++k)
        o += hid[k] * Wn2[k * HDIM + n];
    int oi = (b * NFEAT + n) * NCH + c;
    out[oi] = vnf[oi] + o;   // residual, transposed back to [B,NF,C]
}

// ------------------------------ main kernel --------------------------------
// One wave = one 16-node tile. GEMMs via v_wmma_f32_16x16x4_f32.
// Fragment layouts (ISA 7.12.2, wave32):
//   A 16x4 f32 : lane l16 -> row M=l16; half h -> K = 4*kc + 2h + {0,1}
//   B 4x16 f32 : lane l16 -> col N;     half h -> K = 4*kc + 2h + {0,1}
//   C/D 16x16  : VGPR r -> M = r + 8h, lane l16 -> N

// 16x128 = X(16x128, LDS) @ W(128x128, packed) into 8 accumulators.
// wpg must already include the per-lane (+lane*2) offset.
__device__ __forceinline__ void gemm16(const float* __restrict__ bufX,
                                       const float* __restrict__ wpg,
                                       int l16, int half, v8f acc[8]) {
    #pragma unroll 4
    for (int kc = 0; kc < 32; ++kc) {
        v2f a = *(const v2f*)(bufX + l16 * PITCH + kc * 4 + half * 2);
        #pragma unroll
        for (int j = 0; j < 8; ++j) {
            v2f b = *(const v2f*)(wpg + j * 2048 + kc * 64);
            acc[j] = __builtin_amdgcn_wmma_f32_16x16x4_f32(
                false, a, false, b, (short)0, acc[j], false, false);
        }
    }
}

__global__ void __launch_bounds__(256)
k_main(const float* __restrict__ node_feat,
       const float* __restrict__ coordp,
       const float* __restrict__ vcoord,
       const int*   __restrict__ batch,
       const float* __restrict__ Wp,    // packed [3][8][32][32][2]
       const float* __restrict__ wr,    // We1 radial row [128]
       const float* __restrict__ be2,
       const float* __restrict__ bc1,
       const float* __restrict__ Wc2,   // [128]
       const float* __restrict__ vbias, // [B*C,128]
       float*       __restrict__ aggSum,   // [B*C,128]
       float*       __restrict__ transSum) // [B,3,C]
{
    extern __shared__ float smem[];
    const int wid  = threadIdx.x >> 5;
    const int lane = threadIdx.x & 31;
    const int half = lane >> 4;
    const int l16  = lane & 15;
    const int tile = blockIdx.x * WAVES_PER_BLOCK + wid;
    if (tile >= NTILES) return;
    const int row0 = tile * TILE;

    // per-wave LDS: x/x1 buffer, base buffer, m buffer, small scratch
    float* bufA = smem + wid * WAVE_LDS_FLOATS;  // 16x132
    float* bufB = bufA + TILE * PITCH;           // 16x132 (base = nf @ We1_R)
    float* bufM = bufB + TILE * PITCH;           // 16x132 (m)
    float* sm   = bufM + TILE * PITCH;           // 96 floats
    int*   bb   = (int*)sm;                      // [16] graph id per row
    float* dr   = sm + 16;                       // [16][4] dx,dy,dz,radial
    float* srow = sm + 80;                       // [16] coord scalar s per row

    const float* wpl = Wp + lane * 2;            // per-lane packed-weight base

    // ---- stage node_feat tile into LDS via async global->LDS b128 copies ----
    // (dynamic shared starts at LDS offset 0; no static LDS in this kernel)
    {
        const unsigned wbase = (unsigned)(wid * WAVE_LDS_FLOATS);   // floats
        #pragma unroll
        for (int i = 0; i < 16; ++i) {
            int chunk = i * 32 + lane;             // 512 chunks of 16B
            int row = chunk >> 5;                  // 32 chunks per 128-f row
            int col = (chunk & 31) << 2;           // float index in row
            unsigned ldsb = (wbase + (unsigned)(row * PITCH + col)) * 4u;
            unsigned long long ga = (unsigned long long)(uintptr_t)
                (node_feat + (size_t)(row0 + row) * NFEAT + col);
            asm volatile("global_load_async_to_lds_b128 %0, %1, off"
                         :: "v"(ldsb), "v"(ga) : "memory");
        }
    }

    // per-row metadata (regular path)
    float cox = 0.f, coy = 0.f, coz = 0.f;
    if (lane < 16) {
        int node = row0 + lane;
        bb[lane] = batch[node];
        cox = coordp[node * 3 + 0];
        coy = coordp[node * 3 + 1];
        coz = coordp[node * 3 + 2];
    }
    if (tile + WAVES_PER_BLOCK < NTILES)
        __builtin_prefetch(node_feat + (size_t)(row0 + WAVES_PER_BLOCK * TILE) * NFEAT, 0, 0);

    asm volatile("s_wait_asynccnt 0x0" ::: "memory");
    LDS_FENCE();

    // uniform-batch fast-path check (data_batch is sorted, boundaries rare)
    bool rowok = (lane < 16) ? (bb[lane] == bb[0]) : true;
    unsigned long long vote = __ballot((int)rowok);
    const bool uni = ((vote & 0xffffffffull) == 0xffffffffull);
    const int  b0  = bb[0];

    const v8f zero8 = {0, 0, 0, 0, 0, 0, 0, 0};

    // ---- GEMM1: base = node_feat_tile @ We1[0:128]  (no bias; be1 in vbias)
    {
        v8f acc[8];
        #pragma unroll
        for (int j = 0; j < 8; ++j) acc[j] = zero8;
        gemm16(bufA, wpl + 0, l16, half, acc);
        #pragma unroll
        for (int j = 0; j < 8; ++j) {
            const int col = j * 16 + l16;
            #pragma unroll
            for (int r = 0; r < 8; ++r)
                bufB[(r + half * 8) * PITCH + col] = acc[j][r];
        }
    }
    LDS_FENCE();

    // ---------------------- per-virtual-channel loop -----------------------
    for (int c = 0; c < NCH; ++c) {
        // diff / radial per row
        if (lane < 16) {
            int b = bb[lane];
            float dx = vcoord[(b * 3 + 0) * NCH + c] - cox;
            float dy = vcoord[(b * 3 + 1) * NCH + c] - coy;
            float dz = vcoord[(b * 3 + 2) * NCH + c] - coz;
            dr[lane * 4 + 0] = dx;
            dr[lane * 4 + 1] = dy;
            dr[lane * 4 + 2] = dz;
            dr[lane * 4 + 3] = sqrtf(dx * dx + dy * dy + dz * dz);
        }
        LDS_FENCE();

        // x1 = silu(base + vbias[b,c,:] + radial * wr)  -> bufA
        for (int row = 0; row < 16; ++row) {
            const int   vb  = (bb[row] * NCH + c) * HDIM;
            const float rad = dr[row * 4 + 3];
            for (int col = lane; col < HDIM; col += 32) {
                float x = bufB[row * PITCH + col] + vbias[vb + col] + rad * wr[col];
                bufA[row * PITCH + col] = silu_f(x);
            }
        }
        LDS_FENCE();

        // ---- GEMM2: m = silu(x1 @ We2 + be2) -> bufM, plus segment-sum agg
        {
            v8f acc[8];
            #pragma unroll
            for (int j = 0; j < 8; ++j) acc[j] = zero8;
            gemm16(bufA, wpl + 16384, l16, half, acc);
            #pragma unroll
            for (int j = 0; j < 8; ++j) {
                const int col  = j * 16 + l16;
                const float bias = be2[col];
                float colsum = 0.f;
                #pragma unroll
                for (int r = 0; r < 8; ++r) {
                    float v = silu_f(acc[j][r] + bias);
                    bufM[(r + half * 8) * PITCH + col] = v;
                    colsum += v;
                }
                if (uni) {   // whole tile in one graph: column-reduce + 16 atomics
                    colsum += __shfl_xor(colsum, 16);
                    if (lane < 16)
                        atomicAdd(&aggSum[(b0 * NCH + c) * HDIM + j * 16 + lane], colsum);
                }
            }
        }
        if (!uni) {      // boundary tile: per-element atomics from LDS
            LDS_FENCE();
            for (int e = lane; e < TILE * HDIM; e += 32) {
                int row = e >> 7, col = e & 127;
                atomicAdd(&aggSum[(bb[row] * NCH + c) * HDIM + col],
                          bufM[row * PITCH + col]);
            }
        }
        LDS_FENCE();

        // ---- GEMM3: s = silu(m @ Wc1 + bc1) @ Wc2, fused dot in registers
        float sacc[8] = {0, 0, 0, 0, 0, 0, 0, 0};
        {
            v8f acc[8];
            #pragma unroll
            for (int j = 0; j < 8; ++j) acc[j] = zero8;
            gemm16(bufM, wpl + 32768, l16, half, acc);
            #pragma unroll
            for (int j = 0; j < 8; ++j) {
                const int col  = j * 16 + l16;
                const float bias = bc1[col];
                const float w2   = Wc2[col];
                #pragma unroll
                for (int r = 0; r < 8; ++r)
                    sacc[r] += silu_f(acc[j][r] + bias) * w2;
            }
        }
        // reduce the 16 N-lanes inside each half; row = r + 8*half
        #pragma unroll
        for (int r = 0; r < 8; ++r) {
            float v = sacc[r];
            v += __shfl_xor(v, 1);
            v += __shfl_xor(v, 2);
            v += __shfl_xor(v, 4);
            v += __shfl_xor(v, 8);
            if (l16 == 0) srow[half * 8 + r] = v;
        }
        LDS_FENCE();

        // trans = diff * s -> segment-sum of coordinate update
        if (lane < 16) {
            float s = srow[lane];
            int   b = bb[lane];
            atomicAdd(&transSum[(b * 3 + 0) * NCH + c], dr[lane * 4 + 0] * s);
            atomicAdd(&transSum[(b * 3 + 1) * NCH + c], dr[lane * 4 + 1] * s);
            atomicAdd(&transSum[(b * 3 + 2) * NCH + c], dr[lane * 4 + 2] * s);
        }
        LDS_FENCE();
    }
}

// ------------------------------- launcher ----------------------------------

extern "C" void kernel_launch(void* const* d_in, const int* in_sizes, int n_in,
                              void* d_out, int out_size, void* d_ws, size_t ws_size,
                              hipStream_t stream) {
    const float* node_feat = (const float*)d_in[0];
    const float* coordp    = (const float*)d_in[1];
    const float* vnf       = (const float*)d_in[2];
    const float* vcoord    = (const float*)d_in[3];
    const int*   batch     = (const int*)  d_in[4];
    const float* We1 = (const float*)d_in[5];
    const float* be1 = (const float*)d_in[6];
    const float* We2 = (const float*)d_in[7];
    const float* be2 = (const float*)d_in[8];
    const float* Wc1 = (const float*)d_in[9];
    const float* bc1 = (const float*)d_in[10];
    const float* Wc2 = (const float*)d_in[11];
    const float* Wn1 = (const float*)d_in[12];
    const float* bn1 = (const float*)d_in[13];
    const float* Wn2 = (const float*)d_in[14];
    const float* bn2 = (const float*)d_in[15];

    float* out = (float*)d_out;                  // [B*NF*C] + [B*3*C]
    float* ws  = (float*)d_ws;
    float* vbias    = ws;                        // 38400
    float* aggSum   = ws + 38400;                // 38400
    float* transSum = ws + 76800;                // 900
    float* counts   = ws + 77700;                // 100
    float* Wpack    = ws + 77800;                // 49152 (3 * 128 * 128)

    // zero accumulators (aggSum + transSum + counts = 39400 floats)
    k_zero<<<(39400 + 255) / 256, 256, 0, stream>>>(ws + 38400, 39400);

    // pack weights into WMMA fragment order
    k_pack<<<(3 * 8 * 32 * 32 + 255) / 256, 256, 0, stream>>>(We1, We2, Wc1, Wpack);

    // per-(graph,channel) edge-MLP bias precompute
    k_vbias<<<NGRAPH * NCH, HDIM, 0, stream>>>(vnf, We1, be1, vbias);

    // node counts per graph
    k_counts<<<(NNODES + 255) / 256, 256, 0, stream>>>(batch, counts);

    // main fused WMMA kernel: 12500 tiles, 8 waves/block
    const int nblk = (NTILES + WAVES_PER_BLOCK - 1) / WAVES_PER_BLOCK; // 1563
    const size_t shmem = (size_t)WAVES_PER_BLOCK * WAVE_LDS_FLOATS * sizeof(float);
    k_main<<<nblk, 256, shmem, stream>>>(node_feat, coordp, vcoord, batch,
                                         Wpack, We1 + 256 * HDIM,
                                         be2, bc1, Wc2,
                                         vbias, aggSum, transSum);

    // finalize coordinates
    k_final_coord<<<(NGRAPH * 3 * NCH + 127) / 128, 128, 0, stream>>>(
        vcoord, transSum, counts, out + NGRAPH * NFEAT * NCH);

    // virtual-node MLP + residual
    k_node_mlp<<<NGRAPH * NCH, HDIM, 0, stream>>>(vnf, aggSum, counts,
                                                  Wn1, bn1, Wn2, bn2, out);
}